// CosmosUnpatcher3d_19524921328116
// MI455X (gfx1250) — compile-verified
//
#include <hip/hip_runtime.h>
#include <stdint.h>

// Reference: 3D inverse Haar (one level) on (2, 24, 9, 256, 256) f32,
// channels = 8 subbands x 3 channels; rescale * (1/sqrt2)^3 * sqrt(8) == 1,
// so each output is a +/-1-signed sum of the 8 subband values:
//   out[b,c,2ti+dt,2hi+dh,2wi+dw] = sum_s (-1)^(dt*s2+dh*s1+dw*s0) in[b,3s+c,ti,hi,wi]
// then output frame t=0 is dropped -> (2, 3, 17, 512, 512).
//
// Pure HBM streaming (113 MB in + 107 MB out, ~0.19 FLOP/B): the CDNA5 path
// that matters is GLOBAL_LOAD_ASYNC_TO_LDS_B128 + ASYNCcnt double buffering,
// not WMMA (matrix units add only marshalling overhead for +/-1 sums).

#define NB    2
#define NC    3
#define TIN   9
#define HIN   256
#define WIN   256
#define TOUT  17
#define HOUT  512
#define WOUT  512
#define NSUB  8

#define SSTRIDE ((size_t)NC * TIN * HIN * WIN)   // float stride between subbands
#define TILES_PER_BLOCK 4                        // tiles of 2 H-rows each

__device__ __forceinline__ void async_ld_b128(uint32_t lds_byte_off,
                                              unsigned long long gaddr) {
  // GLOBAL_LOAD_ASYNC_TO_LDS_B128: per-lane 16B global -> LDS, ASYNCcnt-tracked.
  asm volatile("global_load_async_to_lds_b128 %0, %1, off"
               :: "v"(lds_byte_off), "v"(gaddr)
               : "memory");
}

__device__ __forceinline__ float2 f2add(float2 a, float2 b) {
  return make_float2(a.x + b.x, a.y + b.y);
}
__device__ __forceinline__ float2 f2sub(float2 a, float2 b) {
  return make_float2(a.x - b.x, a.y - b.y);
}

__global__ __launch_bounds__(256) void
cosmos_unpatch3d_idwt(const float* __restrict__ in, float* __restrict__ out) {
  // Tile = 2 consecutive H-rows x 8 subbands x 256 W = 16 KB; double buffered.
  __shared__ float tile[2][NSUB][2 * WIN];       // 32 KB

  const int t  = threadIdx.x;                    // 0..255
  const int bb = blockIdx.z / NC;                // batch
  const int cc = blockIdx.z % NC;                // channel
  const int ti = blockIdx.y;                     // input frame
  const int hbase = blockIdx.x * (TILES_PER_BLOCK * 2);

  // Subband-0 base (float index) for this (b, c, ti).
  const size_t base0 = (((size_t)(bb * 24 + cc) * TIN + ti) * HIN) * WIN;

  // Fill buffer `bi` with the 2-row tile starting at hi0.
  // 1024 x 16B chunks, 4 per thread; a wave covers 512B consecutive per subband.
  auto issue_tile = [&](int tileIdx, int bi) {
    const int hi0 = hbase + tileIdx * 2;
#pragma unroll
    for (int k = 0; k < 4; ++k) {
      const int q   = t + k * 256;               // 0..1023
      const int s   = q >> 7;                    // subband (128 chunks each)
      const int rem = q & 127;
      const int h   = rem >> 6;                  // row within tile (0/1)
      const int i16 = rem & 63;                  // 16B chunk within row
      const size_t gidx = base0 + (size_t)s * SSTRIDE
                        + (size_t)(hi0 + h) * WIN + (size_t)i16 * 4;
      const unsigned long long ga = (unsigned long long)(uintptr_t)(in + gidx);
      const uint32_t la =
          (uint32_t)(uintptr_t)(&tile[bi][s][h * WIN + i16 * 4]);
      async_ld_b128(la, ga);
    }
  };

  issue_tile(0, 0);                              // prologue

  const size_t plane = (size_t)HOUT * WOUT;
  const size_t obase = (size_t)(bb * NC + cc) * TOUT;
  const int tA = 2 * ti - 1;                     // dt=0 output frame (-1 dropped)
  const int tB = 2 * ti;                         // dt=1 output frame

  for (int j = 0; j < TILES_PER_BLOCK; ++j) {
    if (j + 1 < TILES_PER_BLOCK) {
      issue_tile(j + 1, (j + 1) & 1);            // prefetch next tile
      // 4 outstanding are the next tile's; this tile done once ASYNCcnt <= 4
      // (async loads complete in order).
      asm volatile("s_wait_asynccnt 0x4" ::: "memory");
    } else {
      asm volatile("s_wait_asynccnt 0x0" ::: "memory");
    }
    __syncthreads();                             // tile j visible to all waves

    const int bi = j & 1;
    const int h  = t >> 7;                       // row within tile (0/1)
    const int wp = t & 127;                      // W-pair index
    const int w0 = 2 * wp;
    const int hi = hbase + j * 2 + h;

    // 8 subbands, two W-columns each: ds_load_b64, conflict-free (8B/lane).
    const float* base = &tile[bi][0][h * WIN + w0];
    const float2 v0 = *(const float2*)(base + 0 * 2 * WIN);
    const float2 v1 = *(const float2*)(base + 1 * 2 * WIN);
    const float2 v2 = *(const float2*)(base + 2 * 2 * WIN);
    const float2 v3 = *(const float2*)(base + 3 * 2 * WIN);
    const float2 v4 = *(const float2*)(base + 4 * 2 * WIN);
    const float2 v5 = *(const float2*)(base + 5 * 2 * WIN);
    const float2 v6 = *(const float2*)(base + 6 * 2 * WIN);
    const float2 v7 = *(const float2*)(base + 7 * 2 * WIN);

    // 3-stage +/- butterfly (W, H, T); .x/.y are the two independent columns.
    const float2 e0 = f2add(v0, v1), o0 = f2sub(v0, v1);
    const float2 e1 = f2add(v2, v3), o1 = f2sub(v2, v3);
    const float2 e2 = f2add(v4, v5), o2 = f2sub(v4, v5);
    const float2 e3 = f2add(v6, v7), o3 = f2sub(v6, v7);
    const float2 le0 = f2add(e0, e1), le1 = f2sub(e0, e1);
    const float2 lo0 = f2add(o0, o1), lo1 = f2sub(o0, o1);
    const float2 he0 = f2add(e2, e3), he1 = f2sub(e2, e3);
    const float2 ho0 = f2add(o2, o3), ho1 = f2sub(o2, o3);

    // 4 consecutive output W values per (dt, dh): full b128 stores.
    const float4 r00 = make_float4(le0.x + he0.x, lo0.x + ho0.x,
                                   le0.y + he0.y, lo0.y + ho0.y); // dt0, dh0
    const float4 r01 = make_float4(le1.x + he1.x, lo1.x + ho1.x,
                                   le1.y + he1.y, lo1.y + ho1.y); // dt0, dh1
    const float4 r10 = make_float4(le0.x - he0.x, lo0.x - ho0.x,
                                   le0.y - he0.y, lo0.y - ho0.y); // dt1, dh0
    const float4 r11 = make_float4(le1.x - he1.x, lo1.x - ho1.x,
                                   le1.y - he1.y, lo1.y - ho1.y); // dt1, dh1

    const size_t hw0 = (size_t)(2 * hi) * WOUT + (size_t)4 * wp;  // dh=0 row
    const size_t hw1 = hw0 + WOUT;                                // dh=1 row

    if (tA >= 0) {
      float* p = out + (obase + (size_t)tA) * plane;
      *(float4*)(p + hw0) = r00;
      *(float4*)(p + hw1) = r01;
    }
    {
      float* p = out + (obase + (size_t)tB) * plane;
      *(float4*)(p + hw0) = r10;
      *(float4*)(p + hw1) = r11;
    }
    __syncthreads();                             // buffer refilled next iter
  }
}

extern "C" void kernel_launch(void* const* d_in, const int* in_sizes, int n_in,
                              void* d_out, int out_size, void* d_ws, size_t ws_size,
                              hipStream_t stream) {
  (void)in_sizes; (void)n_in; (void)out_size; (void)d_ws; (void)ws_size;
  const float* in = (const float*)d_in[0];
  float* out      = (float*)d_out;

  // grid.x: 256 H-rows / (4 tiles * 2 rows) = 32; grid.y: ti; grid.z: b*c
  dim3 grid(HIN / (TILES_PER_BLOCK * 2), TIN, NB * NC);
  cosmos_unpatch3d_idwt<<<grid, 256, 0, stream>>>(in, out);
}